// MeanSpikeClassifier_31525059952997
// MI455X (gfx1250) — compile-verified
//
#include <hip/hip_runtime.h>
#include <hip/hip_bf16.h>

#define B_      16
#define T_      256
#define N_      2304          // 48*48
#define NP_     4096          // pow2 pad for bitonic sort
#define START_  127           // START_STEP-1
#define BS      256

typedef __attribute__((ext_vector_type(2))) float v2f;
typedef __attribute__((ext_vector_type(8))) float v8f;

// ---------------------------------------------------------------------------
// 1) Time-mean via WMMA: avg[b][n] = (1/129) * sum_{t=127..255} x[b][t][n]
//    A = ones(16x4) f32, B = 4 time rows x 16 pixels, D accumulates in f32.
//    One wave handles 16 pixels; 32 WMMAs cover t=127..254, scalar tail t=255.
// ---------------------------------------------------------------------------
__global__ __launch_bounds__(128)
void mean_wmma_kernel(const float* __restrict__ x, float* __restrict__ avg) {
    int gtid = blockIdx.x * blockDim.x + threadIdx.x;
    int wave = gtid >> 5;
    int lane = threadIdx.x & 31;

    int b    = wave / (N_ / 16);
    int nblk = wave % (N_ / 16);
    int n0   = nblk * 16;
    int np   = n0 + (lane & 15);
    int th   = (lane >> 4) * 2;      // lanes 0-15 -> K 0,1 ; lanes 16-31 -> K 2,3

    const float* base = x + (size_t)b * T_ * N_;

    v2f a; a.x = 1.0f; a.y = 1.0f;   // all-ones A matrix
    v8f c = {};
    for (int it = 0; it < 32; ++it) {
        int t = START_ + it * 4 + th;
        v2f bb;
        bb.x = base[(size_t)t * N_ + np];
        bb.y = base[(size_t)(t + 1) * N_ + np];
        c = __builtin_amdgcn_wmma_f32_16x16x4_f32(
                false, a, false, bb, (short)0, c, false, false);
    }
    float s = c[0];                   // every row of D equals the column sum
    s += base[(size_t)(T_ - 1) * N_ + np];          // tail step t=255
    if (lane < 16)
        avg[(size_t)b * N_ + np] = s * (1.0f / 129.0f);
}

// ---------------------------------------------------------------------------
// 2) Per-batch connected components on the line: sort + gap-scan + labels
// ---------------------------------------------------------------------------
__device__ __forceinline__ void inclusive_scan_2304(int* s, int tid) {
    for (int off = 1; off < N_; off <<= 1) {
        int tmp[9];
#pragma unroll
        for (int k = 0; k < 9; ++k) {
            int i = tid + (k << 8);
            int t = s[i];
            if (i >= off) t += s[i - off];
            tmp[k] = t;
        }
        __syncthreads();
#pragma unroll
        for (int k = 0; k < 9; ++k) s[tid + (k << 8)] = tmp[k];
        __syncthreads();
    }
}

__global__ __launch_bounds__(BS)
void components_kernel(const float* __restrict__ avg, int* __restrict__ rank_out) {
    __shared__ float s_val[NP_];
    __shared__ int   s_idx[NP_];
    __shared__ int   s_comp[N_];
    __shared__ int   s_min[N_];
    __shared__ int   s_label[N_];

    int b   = blockIdx.x;
    int tid = threadIdx.x;
    const float* v = avg + (size_t)b * N_;

    for (int i = tid; i < NP_; i += BS) {
        s_val[i] = (i < N_) ? v[i] : 3.0e38f;   // pad sorts last
        s_idx[i] = i;
    }
    __syncthreads();

    // bitonic sort ascending by value (ties order-irrelevant for components)
    for (int k = 2; k <= NP_; k <<= 1) {
        for (int j = k >> 1; j > 0; j >>= 1) {
            for (int i = tid; i < NP_; i += BS) {
                int l = i ^ j;
                if (l > i) {
                    bool up = ((i & k) == 0);
                    float vi = s_val[i], vl = s_val[l];
                    if (up ? (vi > vl) : (vi < vl)) {
                        s_val[i] = vl; s_val[l] = vi;
                        int ti = s_idx[i]; s_idx[i] = s_idx[l]; s_idx[l] = ti;
                    }
                }
            }
            __syncthreads();
        }
    }

    const float EPSF = (float)(1.0 - 0.9);   // matches reference threshold cast
    for (int i = tid; i < N_; i += BS) {
        s_comp[i] = (i == 0) ? 1 : (((s_val[i] - s_val[i - 1]) > EPSF) ? 1 : 0);
        s_min[i]  = 0x7fffffff;
    }
    __syncthreads();

    inclusive_scan_2304(s_comp, tid);        // comp id = s_comp[i]-1

    for (int i = tid; i < N_; i += BS)
        atomicMin(&s_min[s_comp[i] - 1], s_idx[i]);   // smallest pixel per comp
    __syncthreads();

    for (int i = tid; i < N_; i += BS)
        s_label[s_idx[i]] = s_min[s_comp[i] - 1];     // label per pixel
    __syncthreads();

    for (int i = tid; i < N_; i += BS)
        s_comp[i] = (s_label[i] == i) ? 1 : 0;        // representatives
    __syncthreads();

    inclusive_scan_2304(s_comp, tid);        // dense rank via cumsum of reps

    for (int i = tid; i < N_; i += BS)
        rank_out[(size_t)b * N_ + i] = s_comp[s_label[i]] - 1;
}

// ---------------------------------------------------------------------------
// 3) Output: zero-fill 340MB with B128 stores, then scatter the 36864 ones
// ---------------------------------------------------------------------------
__global__ void zero_kernel(float4* __restrict__ out, int n4) {
    int i      = blockIdx.x * blockDim.x + threadIdx.x;
    int stride = gridDim.x * blockDim.x;
    float4 z; z.x = 0.f; z.y = 0.f; z.z = 0.f; z.w = 0.f;
    for (; i < n4; i += stride) out[i] = z;
}

__global__ void scatter_kernel(const int* __restrict__ rank, float* __restrict__ out) {
    int p = blockIdx.x * blockDim.x + threadIdx.x;
    if (p >= B_ * N_) return;
    int b = p / N_, n = p % N_;
    int r = rank[p];
    out[(size_t)b * N_ * N_ + (size_t)r * N_ + n] = 1.0f;
}

// ---------------------------------------------------------------------------
extern "C" void kernel_launch(void* const* d_in, const int* in_sizes, int n_in,
                              void* d_out, int out_size, void* d_ws, size_t ws_size,
                              hipStream_t stream) {
    const float* x   = (const float*)d_in[0];
    float*       out = (float*)d_out;

    float* d_avg  = (float*)d_ws;                                  // 16*2304 f32
    int*   d_rank = (int*)((char*)d_ws + sizeof(float) * B_ * N_); // 16*2304 i32

    const int total4 = (B_ * N_ * N_) / 4;   // 21,233,664 float4s

    zero_kernel<<<4096, 256, 0, stream>>>((float4*)out, total4);
    mean_wmma_kernel<<<(B_ * (N_ / 16)) / 4, 128, 0, stream>>>(x, d_avg);
    components_kernel<<<B_, BS, 0, stream>>>(d_avg, d_rank);
    scatter_kernel<<<(B_ * N_ + 255) / 256, 256, 0, stream>>>(d_rank, out);
}